// OpenFoldFK_71897752535477
// MI455X (gfx1250) — compile-verified
//
#include <hip/hip_runtime.h>
#include <stdint.h>

// ---------------------------------------------------------------------------
// OpenFold frame-and-atom forward kinematics for MI455X (gfx1250).
// Memory-bound (~51 MB @ 23.3 TB/s => ~2.2us floor). One thread = one residue.
// Residue-type constant tables are staged into LDS with the CDNA5 Tensor Data
// Mover (tensor_load_to_lds + s_wait_tensorcnt), fallback = cooperative copy.
// ---------------------------------------------------------------------------

typedef __attribute__((ext_vector_type(4))) unsigned int u32x4;
typedef __attribute__((ext_vector_type(8))) int          i32x8;
typedef __attribute__((ext_vector_type(4))) int          i32x4;

#ifndef __has_builtin
#define __has_builtin(x) 0
#endif

#if __has_builtin(__builtin_amdgcn_tensor_load_to_lds) && \
    __has_builtin(__builtin_amdgcn_s_wait_tensorcnt)
#define FK_USE_TDM 1
#else
#define FK_USE_TDM 0
#endif

#if FK_USE_TDM
// Build a D# (ISA cdna5 ch.8) for a 1-row tile of `ndw` dwords: global -> LDS.
// This toolchain's builtin is the 6-arg form:
//   (u32x4 group0, i32x8 group1, i32x4 group2, i32x4 group3, i32x8 ext, i32 cpol)
__device__ __forceinline__ void fk_tdm_load(uint32_t lds_byte_addr,
                                            const void* gptr, uint32_t ndw) {
  uint64_t ga = (uint64_t)(uintptr_t)gptr;
  u32x4 g0;
  g0.x = 1u;                                       // count=1, user mode
  g0.y = lds_byte_addr;                            // lds_addr [63:32]
  g0.z = (uint32_t)(ga & 0xFFFFFFFFu);             // global_addr [95:64]
  g0.w = (uint32_t)((ga >> 32) & 0x1FFFFFFu)       // global_addr [120:96]
       | (2u << 30);                               // type=2 ("image")
  i32x8 g1;
  g1[0] = (int)(2u << 16);                         // data_size = 4 bytes
  g1[1] = (int)(ndw << 16);                        // tensor_dim0 [15:0] @ [63:48]
  g1[2] = (int)((ndw >> 16) | (1u << 16));         // tensor_dim0 hi, tensor_dim1=1
  g1[3] = (int)(ndw << 16);                        // tile_dim0 = ndw @ [127:112]
  g1[4] = 1;                                       // tile_dim1 = 1
  g1[5] = (int)ndw;                                // tensor_dim0_stride
  g1[6] = 0;
  g1[7] = 0;
  i32x4 z4 = {0, 0, 0, 0};                         // groups 2/3 unused (2D tile)
  i32x8 z8 = {0, 0, 0, 0, 0, 0, 0, 0};             // extended group, unused
  __builtin_amdgcn_tensor_load_to_lds(g0, g1, z4, z4, z8, 0);
}
#endif

__global__ void __launch_bounds__(256)
fk_kernel(const float* __restrict__ tor,    // [BN,7,2] sin,cos
          const float* __restrict__ rots,   // [BN,3,3]
          const float* __restrict__ trans,  // [BN,3]
          const int*   __restrict__ aatype, // [BN]
          const float* __restrict__ lit,    // [20,14,3]
          const int*   __restrict__ grp,    // [20,14]
          const float* __restrict__ msk,    // [20,14]
          float* __restrict__ out, int BN) {
  __shared__ float s_lit[20 * 14 * 3];
  __shared__ int   s_grp[20 * 14];
  __shared__ float s_msk[20 * 14];

#if FK_USE_TDM
  if (threadIdx.x == 0) {
    fk_tdm_load((uint32_t)(uintptr_t)(void*)s_lit, lit, 20u * 14u * 3u);
    fk_tdm_load((uint32_t)(uintptr_t)(void*)s_grp, grp, 20u * 14u);
    fk_tdm_load((uint32_t)(uintptr_t)(void*)s_msk, msk, 20u * 14u);
    __builtin_amdgcn_s_wait_tensorcnt(0);
  }
#else
  for (int i = threadIdx.x; i < 20 * 14 * 3; i += 256) s_lit[i] = lit[i];
  for (int i = threadIdx.x; i < 20 * 14; i += 256) {
    s_grp[i] = grp[i];
    s_msk[i] = msk[i];
  }
#endif
  __syncthreads();

  const int r = blockIdx.x * 256 + threadIdx.x;
  if (r >= BN) return;

  // Torsion sin/cos for the 8 groups (group 0 = identity).
  float sn[8], cs[8];
  sn[0] = 0.0f;
  cs[0] = 1.0f;
  const float2* t2 = (const float2*)(tor + (size_t)r * 14);
#pragma unroll
  for (int k = 0; k < 7; ++k) {
    float2 v = t2[k];
    sn[k + 1] = v.x;
    cs[k + 1] = v.y;
  }

  const float* Rp = rots + (size_t)r * 9;
  const float R00 = Rp[0], R01 = Rp[1], R02 = Rp[2];
  const float R10 = Rp[3], R11 = Rp[4], R12 = Rp[5];
  const float R20 = Rp[6], R21 = Rp[7], R22 = Rp[8];
  const float* tp = trans + (size_t)r * 3;
  const float tx = tp[0], ty = tp[1], tz = tp[2];

  const int aa = aatype[r];
  const float* litb = &s_lit[aa * 42];
  const int*   grb  = &s_grp[aa * 14];
  const float* mkb  = &s_msk[aa * 14];

  float* po = out + (size_t)r * 42;
  float* mo = out + (size_t)BN * 42 + (size_t)r * 14;

#pragma unroll
  for (int a = 0; a < 14; ++a) {
    const int g = grb[a];
    const float c = cs[g], s = sn[g];

    // Per-group literature translation (g0,g1 -> 0; g2 -> N; g3 -> C; g>=4 -> CB).
    const float tlx = (g == 2) ? -0.525f : (g == 3) ? 1.526f : (g >= 4) ? -0.529f : 0.0f;
    const float tly = (g == 2) ?  1.363f : (g >= 4) ? -0.774f : 0.0f;
    const float tlz = (g >= 4) ? -1.205f : 0.0f;

    // ta = Rbb @ tl + tbb   (matches reference einsum order)
    const float tax = R00 * tlx + R01 * tly + R02 * tlz + tx;
    const float tay = R10 * tlx + R11 * tly + R12 * tlz + ty;
    const float taz = R20 * tlx + R21 * tly + R22 * tlz + tz;

    // Ra = Rbb @ Rotx(theta): col0 = Rbb col0; cols 1/2 mix Rbb cols 1/2.
    const float a01 = R01 * c + R02 * s, a02 = R02 * c - R01 * s;
    const float a11 = R11 * c + R12 * s, a12 = R12 * c - R11 * s;
    const float a21 = R21 * c + R22 * s, a22 = R22 * c - R21 * s;

    const float lx = litb[a * 3 + 0];
    const float ly = litb[a * 3 + 1];
    const float lz = litb[a * 3 + 2];

    po[a * 3 + 0] = R00 * lx + a01 * ly + a02 * lz + tax;
    po[a * 3 + 1] = R10 * lx + a11 * ly + a12 * lz + tay;
    po[a * 3 + 2] = R20 * lx + a21 * ly + a22 * lz + taz;
    mo[a] = mkb[a];
  }
}

extern "C" void kernel_launch(void* const* d_in, const int* in_sizes, int n_in,
                              void* d_out, int out_size, void* d_ws, size_t ws_size,
                              hipStream_t stream) {
  const float* tor    = (const float*)d_in[0];  // [B,N,7,2]
  const float* rots   = (const float*)d_in[1];  // [B,N,3,3]
  const float* trans  = (const float*)d_in[2];  // [B,N,3]
  const int*   aatype = (const int*)d_in[3];    // [B,N]
  const float* lit    = (const float*)d_in[4];  // [20,14,3]
  const int*   grp    = (const int*)d_in[5];    // [20,14]
  const float* msk    = (const float*)d_in[6];  // [20,14]
  float* out = (float*)d_out;

  const int BN = in_sizes[3];  // B*N residues
  const int blocks = (BN + 255) / 256;
  hipLaunchKernelGGL(fk_kernel, dim3(blocks), dim3(256), 0, stream,
                     tor, rots, trans, aatype, lit, grp, msk, out, BN);
}